// DecodeNextToken_58351425683656
// MI455X (gfx1250) — compile-verified
//
#include <hip/hip_runtime.h>
#include <hip/hip_bf16.h>
#include <stdint.h>

// DecodeNextToken for MI455X (gfx1250): repetition penalty + top-p + top-k +
// softmax + Gumbel argmax sampling. One 512-thread (16 wave32) block per row.
// Memory-bound: ~313 MB mandatory HBM traffic -> ~13.4us floor @ 23.3 TB/s.
// Logits (103 MB) stay resident in the 192 MB L2 across the 4 selection passes.
// All bulk traffic uses 128-bit global ops with a per-row alignment prologue
// (V % 4 == 1 shifts each row's phase by row mod 4).

#define B_ROWS   512
#define VOCAB    50257
#define TPREV    2048
#define TOPK     50
#define NT       512
#define NWAVES   (NT / 32)
#define MASKW    ((VOCAB + 31) / 32)
#define CAND_MAX 1536

typedef __attribute__((ext_vector_type(4))) unsigned int v4u;
typedef __attribute__((ext_vector_type(8))) int          v8i;
typedef __attribute__((ext_vector_type(4))) int          v4i;

struct BlkShared {
  int      prevTok[TPREV];     // TDM destination; LDS offset 0, aligned
  unsigned mask[MASKW + 1];    // +1 pad word so 64-bit windows never read OOB
  unsigned hist1[256];
  unsigned hist2[256];
  float    cand[CAND_MAX];
  float    sortedTop[TOPK + 2];
  float    redF[NWAVES];
  float    redS[NWAVES];
  int      redI[NWAVES];
  unsigned candCount;
  float    rowMax;
  float    rowZ;
  unsigned b1, cHi, tau;
  float    pivot, M2, invZ2;
};

// order-preserving float -> uint key (bigger key == bigger float)
__device__ __forceinline__ unsigned fkey(float f) {
  unsigned u = __float_as_uint(f);
  return (u & 0x80000000u) ? ~u : (u | 0x80000000u);
}

__device__ __forceinline__ float penal1(const unsigned* __restrict__ mask,
                                        float l, int i, float rp, float inv_rp) {
  unsigned hit = (mask[i >> 5] >> (i & 31)) & 1u;
  if (hit) l = (l < 0.0f) ? (l * rp) : (l * inv_rp);
  return l;
}

// penalize 4 consecutive elements starting at `base` (any alignment);
// one 64-bit LDS window supplies all 4 membership bits.
__device__ __forceinline__ float4 penal4(const unsigned* __restrict__ mask,
                                         float4 v, int base, float rp, float inv_rp) {
  int w = base >> 5;
  unsigned long long mw =
      ((unsigned long long)mask[w + 1] << 32) | (unsigned long long)mask[w];
  unsigned bits = (unsigned)(mw >> (base & 31)) & 0xFu;
  if (bits & 1u) v.x = (v.x < 0.0f) ? v.x * rp : v.x * inv_rp;
  if (bits & 2u) v.y = (v.y < 0.0f) ? v.y * rp : v.y * inv_rp;
  if (bits & 4u) v.z = (v.z < 0.0f) ? v.z * rp : v.z * inv_rp;
  if (bits & 8u) v.w = (v.w < 0.0f) ? v.w * rp : v.w * inv_rp;
  return v;
}

// wave32 ballot-aggregated LDS histogram increment (match-any emulation):
// normal logits hit only ~a dozen exponent bins, so raw per-lane LDS atomics
// would serialize 32-way; this cuts atomic count by ~32x.
__device__ __forceinline__ void histAddAgg(unsigned* hist, unsigned bin) {
  bool done = false;
  for (;;) {
    unsigned long long vote = __ballot(!done);
    if (!vote) break;
    int leader = (int)__ffsll(vote) - 1;
    unsigned lbin = (unsigned)__shfl((int)bin, leader, 32);
    unsigned long long same = __ballot(!done && (bin == lbin));
    if ((int)(threadIdx.x & 31) == leader)
      atomicAdd(&hist[lbin], (unsigned)__popcll(same));
    if (bin == lbin) done = true;
  }
}

__global__ __launch_bounds__(NT, 1) void
decode_sample_gfx1250(const float* __restrict__ logits,
                      const int* __restrict__ prev_tokens,
                      const float* __restrict__ rand_u,
                      const float* __restrict__ temperature,
                      const float* __restrict__ top_p_ptr,
                      const float* __restrict__ rep_pen,
                      long long* __restrict__ idx_out,
                      float* __restrict__ probs_out) {
  __shared__ BlkShared sb;
  const int row  = blockIdx.x;
  const int tid  = threadIdx.x;
  const int lane = tid & 31;
  const int wid  = tid >> 5;

  const float rp     = rep_pen[0];
  const float inv_rp = 1.0f / rp;
  const float topp   = top_p_ptr[0];
  const float invT   = 1.0f / fmaxf(temperature[0], 1e-5f);

  // per-row vector phase: (row*V) % 4 == row % 4 because V % 4 == 1
  const int pre       = (4 - (row & 3)) & 3;
  const int nvec      = (VOCAB - pre) >> 2;
  const int tailStart = pre + (nvec << 2);

  for (int i = tid; i < MASKW + 1; i += NT) sb.mask[i] = 0u;
  for (int i = tid; i < 256; i += NT) { sb.hist1[i] = 0u; sb.hist2[i] = 0u; }
  if (tid == 0) sb.candCount = 0u;

  // ---- Phase 1: previous_tokens row -> LDS via Tensor Data Mover ----------
  const int* prow = prev_tokens + (size_t)row * TPREV;
  if (tid < 32) {  // one wave issues the DMA; TENSORcnt is per-wave
    uint64_t ga = (uint64_t)(uintptr_t)prow;
    // generic pointers to LDS carry the LDS byte offset in addr[31:0]
    unsigned lds_off = (unsigned)(uintptr_t)(&sb.prevTok[0]);
    v4u g0;
    g0.x = 1u;                                           // count = 1 descriptor
    g0.y = lds_off;                                      // lds_addr
    g0.z = (unsigned)(ga & 0xFFFFFFFFull);               // global_addr[31:0]
    g0.w = (unsigned)((ga >> 32) & 0x01FFFFFFull)        // global_addr[56:32]
           | 0x80000000u;                                // type = 2 ("image")
    v8i g1;
    g1[0] = 0x00020000;                                  // data_size = 4 bytes
    g1[1] = (int)(((unsigned)TPREV & 0xFFFFu) << 16);    // tensor_dim0 lo16
    g1[2] = (int)(((unsigned)TPREV >> 16) | (1u << 16)); // dim0 hi16 | dim1=1
    g1[3] = (int)(((unsigned)TPREV & 0xFFFFu) << 16);    // tile_dim0 = 2048
    g1[4] = 1;                                           // tile_dim1=1, dim2=0
    g1[5] = (int)TPREV;                                  // dim0_stride lo32
    g1[6] = (int)(((unsigned)TPREV & 0xFFFFu) << 16);    // stride0 hi|stride1 lo
    g1[7] = 0;
    v4i z4 = {0, 0, 0, 0};
#if __clang_major__ >= 23
    v8i z8 = {0, 0, 0, 0, 0, 0, 0, 0};
    __builtin_amdgcn_tensor_load_to_lds(g0, g1, z4, z4, z8, 0);
#else
    __builtin_amdgcn_tensor_load_to_lds(g0, g1, z4, z4, 0);
#endif
    __builtin_amdgcn_s_wait_tensorcnt(0);
  }
  __syncthreads();

  // ---- Phase 2: membership bitmask (duplicates idempotent, matches ref) ---
  for (int i = tid; i < TPREV; i += NT) {
    unsigned tok = (unsigned)sb.prevTok[i];
    if (tok < (unsigned)VOCAB) atomicOr(&sb.mask[tok >> 5], 1u << (tok & 31));
  }
  __syncthreads();

  const float* lrow = logits + (size_t)row * VOCAB;

  // ---- Pass A (HBM): row max + level-1 radix histogram --------------------
  float tmax = -3.0e38f;
  for (int i = tid; i < pre; i += NT) {
    float l = penal1(sb.mask, lrow[i], i, rp, inv_rp);
    tmax = fmaxf(tmax, l);
    histAddAgg(sb.hist1, fkey(l) >> 24);
  }
  for (int j = tid; j < nvec; j += NT) {
    int base = pre + (j << 2);
    __builtin_prefetch(lrow + base + (NT << 2), 0, 0);
    float4 v = *(const float4*)(lrow + base);
    v = penal4(sb.mask, v, base, rp, inv_rp);
    tmax = fmaxf(tmax, fmaxf(fmaxf(v.x, v.y), fmaxf(v.z, v.w)));
    histAddAgg(sb.hist1, fkey(v.x) >> 24);
    histAddAgg(sb.hist1, fkey(v.y) >> 24);
    histAddAgg(sb.hist1, fkey(v.z) >> 24);
    histAddAgg(sb.hist1, fkey(v.w) >> 24);
  }
  for (int i = tailStart + tid; i < VOCAB; i += NT) {
    float l = penal1(sb.mask, lrow[i], i, rp, inv_rp);
    tmax = fmaxf(tmax, l);
    histAddAgg(sb.hist1, fkey(l) >> 24);
  }
  for (int o = 16; o; o >>= 1) tmax = fmaxf(tmax, __shfl_xor(tmax, o, 32));
  if (lane == 0) sb.redF[wid] = tmax;
  __syncthreads();
  if (tid == 0) {
    float m = sb.redF[0];
    for (int w = 1; w < NWAVES; ++w) m = fmaxf(m, sb.redF[w]);
    sb.rowMax = m;
    unsigned cum = 0, b = 0, chi = 0;
    for (int bb = 255; bb >= 0; --bb) {
      unsigned c = sb.hist1[bb];
      if (cum + c >= TOPK) { b = (unsigned)bb; chi = cum; break; }
      cum += c;
    }
    sb.b1 = b; sb.cHi = chi;
  }
  __syncthreads();
  const float    m  = sb.rowMax;
  const unsigned b1 = sb.b1;

  // ---- Pass B (L2): softmax denominator Z + level-2 histogram -------------
  float tz = 0.0f;
  for (int i = tid; i < pre; i += NT) {
    float l = penal1(sb.mask, lrow[i], i, rp, inv_rp);
    tz += __expf(l - m);
    unsigned k = fkey(l);
    if ((k >> 24) == b1) atomicAdd(&sb.hist2[(k >> 16) & 0xFFu], 1u);
  }
  for (int j = tid; j < nvec; j += NT) {
    int base = pre + (j << 2);
    float4 v = *(const float4*)(lrow + base);
    v = penal4(sb.mask, v, base, rp, inv_rp);
    tz += __expf(v.x - m) + __expf(v.y - m) + __expf(v.z - m) + __expf(v.w - m);
    unsigned k0 = fkey(v.x), k1 = fkey(v.y), k2 = fkey(v.z), k3 = fkey(v.w);
    if ((k0 >> 24) == b1) atomicAdd(&sb.hist2[(k0 >> 16) & 0xFFu], 1u);
    if ((k1 >> 24) == b1) atomicAdd(&sb.hist2[(k1 >> 16) & 0xFFu], 1u);
    if ((k2 >> 24) == b1) atomicAdd(&sb.hist2[(k2 >> 16) & 0xFFu], 1u);
    if ((k3 >> 24) == b1) atomicAdd(&sb.hist2[(k3 >> 16) & 0xFFu], 1u);
  }
  for (int i = tailStart + tid; i < VOCAB; i += NT) {
    float l = penal1(sb.mask, lrow[i], i, rp, inv_rp);
    tz += __expf(l - m);
    unsigned k = fkey(l);
    if ((k >> 24) == b1) atomicAdd(&sb.hist2[(k >> 16) & 0xFFu], 1u);
  }
  for (int o = 16; o; o >>= 1) tz += __shfl_xor(tz, o, 32);
  if (lane == 0) sb.redF[wid] = tz;
  __syncthreads();
  if (tid == 0) {
    float Z = 0.0f;
    for (int w = 0; w < NWAVES; ++w) Z += sb.redF[w];
    sb.rowZ = Z;
    unsigned cum = sb.cHi, b2 = 0;
    for (int bb = 255; bb >= 0; --bb) {
      unsigned c = sb.hist2[bb];
      if (cum + c >= TOPK) { b2 = (unsigned)bb; break; }
      cum += c;
    }
    sb.tau = (b1 << 24) | (b2 << 16);
  }
  __syncthreads();
  const unsigned tau = sb.tau;

  // ---- Pass C (L2): gather candidates >= tau ------------------------------
  for (int i = tid; i < pre; i += NT) {
    float l = penal1(sb.mask, lrow[i], i, rp, inv_rp);
    if (fkey(l) >= tau) {
      unsigned p = atomicAdd(&sb.candCount, 1u);
      if (p < CAND_MAX) sb.cand[p] = l;
    }
  }
  for (int j = tid; j < nvec; j += NT) {
    int base = pre + (j << 2);
    float4 v = *(const float4*)(lrow + base);
    v = penal4(sb.mask, v, base, rp, inv_rp);
    float e[4] = {v.x, v.y, v.z, v.w};
#pragma unroll
    for (int c = 0; c < 4; ++c) {
      if (fkey(e[c]) >= tau) {
        unsigned p = atomicAdd(&sb.candCount, 1u);
        if (p < CAND_MAX) sb.cand[p] = e[c];
      }
    }
  }
  for (int i = tailStart + tid; i < VOCAB; i += NT) {
    float l = penal1(sb.mask, lrow[i], i, rp, inv_rp);
    if (fkey(l) >= tau) {
      unsigned p = atomicAdd(&sb.candCount, 1u);
      if (p < CAND_MAX) sb.cand[p] = l;
    }
  }
  __syncthreads();
  const int C = (sb.candCount < (unsigned)CAND_MAX) ? (int)sb.candCount : CAND_MAX;

  // ---- rank-select the TOPK largest candidates into sorted order ----------
  for (int j = tid; j < C; j += NT) {
    unsigned kj = fkey(sb.cand[j]);
    int r = 0;
    for (int t = 0; t < C; ++t) {
      unsigned kt = fkey(sb.cand[t]);
      r += (kt > kj) || (kt == kj && t < j);   // total order -> unique ranks
    }
    if (r < TOPK) sb.sortedTop[r] = sb.cand[j];
  }
  __syncthreads();

  // ---- thread 0: exact top-p crossing + final pivot/softmax constants -----
  if (tid == 0) {
    int nTop = (C < TOPK) ? C : TOPK;
    float Z = sb.rowZ;
    float cum = 0.0f;
    int kp = 1;                                  // rank 0 always kept
    for (int r = 0; r < nTop; ++r) {
      cum += __expf(sb.sortedTop[r] - m) / Z;
      if (r > 0) { if (cum <= topp) kp = r + 1; else break; }
    }
    int k = kp;                                  // == min(k_p, TOPK)
    float pivot = sb.sortedTop[k - 1];           // pre-temperature pivot
    float M2 = sb.sortedTop[0] * invT;
    float Z2 = 0.0f;
    for (int r = 0; r < k; ++r) Z2 += __expf(sb.sortedTop[r] * invT - M2);
    sb.pivot = pivot;
    sb.M2 = M2;
    sb.invZ2 = 1.0f / Z2;
  }
  __syncthreads();

  const float pivot = sb.pivot, M2 = sb.M2, invZ2 = sb.invZ2;
  const float* urow = rand_u + (size_t)row * VOCAB;
  float* orow = probs_out + (size_t)row * VOCAB;

  // ---- Pass D (HBM in, HBM out): probs + Gumbel argmax sample -------------
  float bestS = -1.0f;
  int   bestI = 0x7FFFFFFF;
  for (int i = tid; i < pre; i += NT) {
    float l = penal1(sb.mask, lrow[i], i, rp, inv_rp);
    float p = (l >= pivot) ? __expf(l * invT - M2) * invZ2 : 0.0f;
    orow[i] = p;
    float s = p / (-__logf(urow[i]));
    if (s > bestS || (s == bestS && i < bestI)) { bestS = s; bestI = i; }
  }
  for (int j = tid; j < nvec; j += NT) {
    int base = pre + (j << 2);
    __builtin_prefetch(urow + base + (NT << 2), 0, 0);
    float4 v = *(const float4*)(lrow + base);
    v = penal4(sb.mask, v, base, rp, inv_rp);
    float4 u = *(const float4*)(urow + base);
    float4 p;
    p.x = (v.x >= pivot) ? __expf(v.x * invT - M2) * invZ2 : 0.0f;
    p.y = (v.y >= pivot) ? __expf(v.y * invT - M2) * invZ2 : 0.0f;
    p.z = (v.z >= pivot) ? __expf(v.z * invT - M2) * invZ2 : 0.0f;
    p.w = (v.w >= pivot) ? __expf(v.w * invT - M2) * invZ2 : 0.0f;
    *(float4*)(orow + base) = p;
    float pc[4] = {p.x, p.y, p.z, p.w};
    float uc[4] = {u.x, u.y, u.z, u.w};
#pragma unroll
    for (int c = 0; c < 4; ++c) {
      float s = pc[c] / (-__logf(uc[c]));
      int idx = base + c;
      if (s > bestS || (s == bestS && idx < bestI)) { bestS = s; bestI = idx; }
    }
  }
  for (int i = tailStart + tid; i < VOCAB; i += NT) {
    float l = penal1(sb.mask, lrow[i], i, rp, inv_rp);
    float p = (l >= pivot) ? __expf(l * invT - M2) * invZ2 : 0.0f;
    orow[i] = p;
    float s = p / (-__logf(urow[i]));
    if (s > bestS || (s == bestS && i < bestI)) { bestS = s; bestI = i; }
  }
  for (int o = 16; o; o >>= 1) {
    float os = __shfl_xor(bestS, o, 32);
    int   oi = __shfl_xor(bestI, o, 32);
    if (os > bestS || (os == bestS && oi < bestI)) { bestS = os; bestI = oi; }
  }
  if (lane == 0) { sb.redS[wid] = bestS; sb.redI[wid] = bestI; }
  __syncthreads();
  if (tid == 0) {
    float bs = sb.redS[0];
    int   bi = sb.redI[0];
    for (int w = 1; w < NWAVES; ++w) {
      float os = sb.redS[w];
      int   oi = sb.redI[w];
      if (os > bs || (os == bs && oi < bi)) { bs = os; bi = oi; }
    }
    idx_out[row] = (long long)bi;
  }
}

extern "C" void kernel_launch(void* const* d_in, const int* in_sizes, int n_in,
                              void* d_out, int out_size, void* d_ws, size_t ws_size,
                              hipStream_t stream) {
  (void)in_sizes; (void)n_in; (void)out_size; (void)d_ws; (void)ws_size;
  const float* logits = (const float*)d_in[0];
  const int*   prev   = (const int*)d_in[1];
  const float* randu  = (const float*)d_in[2];
  const float* temp   = (const float*)d_in[3];
  const float* topp   = (const float*)d_in[4];
  const float* rp     = (const float*)d_in[5];
  // Outputs concatenated flat in return order: idx_next (B x int64), probs (B*V f32)
  long long* idx_out  = (long long*)d_out;
  float* probs_out    = (float*)((char*)d_out + (size_t)B_ROWS * sizeof(long long));
  hipLaunchKernelGGL(decode_sample_gfx1250, dim3(B_ROWS), dim3(NT), 0, stream,
                     logits, prev, randu, temp, topp, rp, idx_out, probs_out);
}